// SequentialVAE_43731357008219
// MI455X (gfx1250) — compile-verified
//
#include <hip/hip_runtime.h>

typedef __attribute__((ext_vector_type(2))) float v2f;
typedef __attribute__((ext_vector_type(8))) float v8f;

#define T_STEPS 8192
#define DXDIM   1024
#define DZDIM   16

__device__ __forceinline__ v8f wmma_f32(v2f a, v2f b, v8f c) {
  // D = A(16x4) x B(4x16) + C, fp32 matrix pipe
  return __builtin_amdgcn_wmma_f32_16x16x4_f32(false, a, false, b, (short)0, c, false, false);
}

// Broadcast lane `l`'s value to all lanes as an SGPR (v_readlane_b32, no LDS round-trip).
// All call sites use compile-time-constant l.
__device__ __forceinline__ float bcast(float v, int l) {
  return __uint_as_float((unsigned)__builtin_amdgcn_readlane((int)__float_as_uint(v), l));
}

// Generic fp32 WMMA GEMM: out = act( A @ W + bias ), A row-major MxK, W row-major KxN.
// TWO_PHASE: out = A0 @ W[0:K] + A1 @ W[K:2K] + bias   (for B = h[1:]@WBtop + h[:-1]@WBbot)
// ADD_DIAG: bias[col] += (col/16 == col%16)            (folds the +I on the A matrices)
template<int WTM, int WTN, int BWM, int BWN, bool RELU, bool ADD_DIAG, bool TWO_PHASE>
__global__ __launch_bounds__(BWM * BWN * 32)
void wmma_gemm(const float* __restrict__ A0, const float* __restrict__ A1,
               const float* __restrict__ W, const float* __restrict__ bias,
               float* __restrict__ out, int Mvalid, int K, int N)
{
  const int lane = threadIdx.x & 31;
  const int wave = threadIdx.x >> 5;
  const int lr = lane & 15;   // row (A) / col (B,C)
  const int hh = lane >> 4;   // K half select
  const int wm = wave % BWM;
  const int wn = wave / BWM;
  const int row0 = blockIdx.x * (BWM * WTM * 16) + wm * (WTM * 16);
  const int col0 = blockIdx.y * (BWN * WTN * 16) + wn * (WTN * 16);
  const int Mclamp = Mvalid - 1;

  v8f acc[WTM][WTN];
  #pragma unroll
  for (int im = 0; im < WTM; ++im)
    #pragma unroll
    for (int in = 0; in < WTN; ++in)
      #pragma unroll
      for (int v = 0; v < 8; ++v) acc[im][in][v] = 0.f;

  const int nPhase = TWO_PHASE ? 2 : 1;
  for (int ph = 0; ph < nPhase; ++ph) {
    const float* __restrict__ Ap = (TWO_PHASE && ph) ? A1 : A0;
    const float* __restrict__ Wp = W + (size_t)ph * K * N;
    for (int k = 0; k < K; k += 4) {
      v2f afrag[WTM];
      #pragma unroll
      for (int im = 0; im < WTM; ++im) {
        int r = row0 + im * 16 + lr; if (r > Mclamp) r = Mclamp;
        afrag[im] = *(const v2f*)(Ap + (size_t)r * K + k + 2 * hh);
      }
      v2f bfrag[WTN];
      #pragma unroll
      for (int in = 0; in < WTN; ++in) {
        int ccol = col0 + in * 16 + lr;
        const float* wptr = Wp + (size_t)(k + 2 * hh) * N + ccol;
        v2f bf; bf[0] = wptr[0]; bf[1] = wptr[N];
        bfrag[in] = bf;
      }
      #pragma unroll
      for (int im = 0; im < WTM; ++im)
        #pragma unroll
        for (int in = 0; in < WTN; ++in)
          acc[im][in] = wmma_f32(afrag[im], bfrag[in], acc[im][in]);
    }
  }

  #pragma unroll
  for (int in = 0; in < WTN; ++in) {
    int ccol = col0 + in * 16 + lr;
    float b = bias[ccol];
    if (ADD_DIAG) b += ((ccol >> 4) == (ccol & 15)) ? 1.0f : 0.0f;
    #pragma unroll
    for (int im = 0; im < WTM; ++im) {
      #pragma unroll
      for (int v = 0; v < 8; ++v) {
        int r = row0 + im * 16 + v + 8 * hh;   // C/D layout: M = v + 8*(lane>=16), N = lane&15
        float val = acc[im][in][v] + b;
        if (RELU) val = fmaxf(val, 0.f);
        if (r < Mvalid) out[(size_t)r * N + ccol] = val;
      }
    }
  }
}

// One wave per timestep t:
//   AA[t] = A[t]A[t]^T + (t>0 ? B[t-1]B[t-1]^T : 0) + 1e-6 I
//   BB[t] = B[t]A[t]^T            (t < T-1)
// A row-major 16x16 block loaded as float2 per lane serves as BOTH the A-operand of M
// and the B-operand of M^T (the two WMMA operand layouts are mirror images).
__global__ __launch_bounds__(256)
void psd_kernel(const float* __restrict__ Afull, const float* __restrict__ Bfull,
                float* __restrict__ AA, float* __restrict__ BB, int T)
{
  const int lane = threadIdx.x & 31;
  const int wave = threadIdx.x >> 5;
  const int t = blockIdx.x * 8 + wave;
  const int lr = lane & 15;
  const int hh = lane >> 4;

  const float* Ab = Afull + (size_t)t * 256;
  v2f fa[4], fb[4], fbp[4];
  #pragma unroll
  for (int kk = 0; kk < 4; ++kk)
    fa[kk] = *(const v2f*)(Ab + lr * 16 + kk * 4 + 2 * hh);

  const int tb = (t < T - 1) ? t : (T - 2);
  const float* Bb = Bfull + (size_t)tb * 256;
  #pragma unroll
  for (int kk = 0; kk < 4; ++kk)
    fb[kk] = *(const v2f*)(Bb + lr * 16 + kk * 4 + 2 * hh);

  const int tp = (t > 0) ? (t - 1) : 0;
  const float* Bp = Bfull + (size_t)tp * 256;
  #pragma unroll
  for (int kk = 0; kk < 4; ++kk) {
    v2f v = *(const v2f*)(Bp + lr * 16 + kk * 4 + 2 * hh);
    if (t == 0) { v[0] = 0.f; v[1] = 0.f; }
    fbp[kk] = v;
  }

  v8f acc1;
  #pragma unroll
  for (int v = 0; v < 8; ++v) acc1[v] = ((v + 8 * hh) == lr) ? 1e-6f : 0.f;
  #pragma unroll
  for (int kk = 0; kk < 4; ++kk) acc1 = wmma_f32(fa[kk], fa[kk], acc1);     // A A^T
  #pragma unroll
  for (int kk = 0; kk < 4; ++kk) acc1 = wmma_f32(fbp[kk], fbp[kk], acc1);   // + Bprev Bprev^T

  v8f acc2;
  #pragma unroll
  for (int v = 0; v < 8; ++v) acc2[v] = 0.f;
  #pragma unroll
  for (int kk = 0; kk < 4; ++kk) acc2 = wmma_f32(fb[kk], fa[kk], acc2);     // B A^T

  float* AAo = AA + (size_t)t * 256;
  #pragma unroll
  for (int v = 0; v < 8; ++v) AAo[(v + 8 * hh) * 16 + lr] = acc1[v];
  if (t < T - 1) {
    float* BBo = BB + (size_t)t * 256;
    #pragma unroll
    for (int v = 0; v < 8; ++v) BBo[(v + 8 * hh) * 16 + lr] = acc2[v];
  }
}

// Sequential block-tridiagonal Cholesky + forward substitution + outputs.
// One wave; lane c owns row c of L/C (registers) and column c of the running solve x.
// All cross-lane matrix elements are v_readlane SGPR broadcasts (compile-time lane ids);
// loads for step t+1 are issued before the math of step t (software pipelining).
__global__ __launch_bounds__(32)
void scan_kernel(const float* __restrict__ AA, const float* __restrict__ BB,
                 const float* __restrict__ mu, const float* __restrict__ eps,
                 float* __restrict__ out, int T)
{
  const int tid = threadIdx.x;
  const int c = tid & 15;   // lanes 16..31 mirror lanes 0..15

  float Lrow[16], invd[16], xcol[16];
  #pragma unroll
  for (int i = 0; i < 16; ++i) { Lrow[i] = 0.f; invd[i] = 0.f; xcol[i] = 0.f; }
  float logacc = 0.f;

  // preload step 0
  float Arow[16], Brow[16];
  {
    const float* ap = AA + c * 16;
    #pragma unroll
    for (int i = 0; i < 16; ++i) Arow[i] = ap[i];
    #pragma unroll
    for (int i = 0; i < 16; ++i) Brow[i] = 0.f;   // unused at t=0
  }
  float evC = eps[c];
  float muC = mu[c];

  for (int t = 0; t < T; ++t) {
    // ---- prefetch step t+1 (clamped; hidden behind this step's dependence chain) ----
    const int tn = (t + 1 < T) ? (t + 1) : t;
    const int bidx = (t < T - 1) ? t : (T - 2);   // Brow for step t+1 is BB[t]
    float ArowN[16], BrowN[16];
    {
      const float* apn = AA + (size_t)tn * 256 + c * 16;
      const float* bpn = BB + (size_t)bidx * 256 + c * 16;
      #pragma unroll
      for (int i = 0; i < 16; ++i) ArowN[i] = apn[i];
      #pragma unroll
      for (int i = 0; i < 16; ++i) BrowN[i] = bpn[i];
    }
    const float evN = eps[(size_t)tn * 16 + c];
    const float muN = mu[(size_t)tn * 16 + c];

    // ---- C = B_{t-1} @ inv(Lprev)^T : lane r forward-substitutes its row of B ----
    float Crow[16];
    if (t > 0) {
      #pragma unroll
      for (int i = 0; i < 16; ++i) {
        float s = Brow[i];
        #pragma unroll
        for (int j = 0; j < i; ++j) s -= bcast(Lrow[j], i) * Crow[j];
        Crow[i] = s * invd[i];
      }
    } else {
      #pragma unroll
      for (int i = 0; i < 16; ++i) Crow[i] = 0.f;
    }

    // ---- M = AA[t] - C C^T (lane c computes row c) ----
    float Mrow[16];
    #pragma unroll
    for (int k = 0; k < 16; ++k) {
      float s = Arow[k];
      #pragma unroll
      for (int j = 0; j < 16; ++j) s -= Crow[j] * bcast(Crow[j], k);
      Mrow[k] = s;
    }

    // ---- L = chol(M); v_rsq on the diagonal keeps the critical chain short ----
    #pragma unroll
    for (int j = 0; j < 16; ++j) {
      float s = Mrow[j];
      #pragma unroll
      for (int k = 0; k < j; ++k) s -= Lrow[k] * bcast(Lrow[k], j);
      float dj = bcast(s, j);
      float inv = __builtin_amdgcn_rsqf(dj);
      float sd = dj * inv;
      invd[j] = inv;
      float v = (c == j) ? sd : s * inv;
      Lrow[j] = (c >= j) ? v : 0.f;
    }

    // ---- x_t = solve(L, I - C x_{t-1}); lane c owns column c ----
    float xn[16];
    #pragma unroll
    for (int i = 0; i < 16; ++i) {
      float s = (c == i) ? 1.f : 0.f;
      #pragma unroll
      for (int j = 0; j < 16; ++j) s -= bcast(Crow[j], i) * xcol[j];
      #pragma unroll
      for (int j = 0; j < i; ++j) s -= bcast(Lrow[j], i) * xn[j];
      xn[i] = s * invd[i];
    }
    #pragma unroll
    for (int i = 0; i < 16; ++i) xcol[i] = xn[i];

    // ---- outputs: r = max(x, 1e-5) elementwise (recurrence keeps the raw x) ----
    float diag = 0.f;
    #pragma unroll
    for (int i = 0; i < 16; ++i) diag = (c == i) ? xn[i] : diag;
    logacc += logf(fmaxf(diag, 1e-5f));

    float z = muC;
    #pragma unroll
    for (int j = 0; j < 16; ++j) z += fmaxf(xn[j], 1e-5f) * bcast(evC, j);
    if (tid < 16) out[1 + (size_t)t * 16 + c] = z;   // sample_z[t][c]

    // rotate pipeline registers
    #pragma unroll
    for (int i = 0; i < 16; ++i) { Arow[i] = ArowN[i]; Brow[i] = BrowN[i]; }
    evC = evN; muC = muN;
  }

  float total = 0.f;
  #pragma unroll
  for (int i = 0; i < 16; ++i) total += bcast(logacc, i);
  if (tid == 0) {
    // entropy = -sum(log diag) + DX*T/2*(1+log(2*pi))
    out[0] = -total + (float)(4194304.0 * 2.8378770664093453);
  }
}

extern "C" void kernel_launch(void* const* d_in, const int* in_sizes, int n_in,
                              void* d_out, int out_size, void* d_ws, size_t ws_size,
                              hipStream_t stream) {
  (void)in_sizes; (void)n_in; (void)out_size; (void)ws_size;
  const float* X   = (const float*)d_in[0];
  const float* W1  = (const float*)d_in[1];
  const float* b1  = (const float*)d_in[2];
  const float* Wmu = (const float*)d_in[3];
  const float* bmu = (const float*)d_in[4];
  const float* WA  = (const float*)d_in[5];
  const float* bA  = (const float*)d_in[6];
  const float* WB  = (const float*)d_in[7];
  const float* bB  = (const float*)d_in[8];
  const float* eps = (const float*)d_in[9];
  float* out = (float*)d_out;

  // Workspace layout (floats): h | A | B | AA | BB | mu  (~64.5 MB total)
  float* h   = (float*)d_ws;
  float* Af  = h   + (size_t)T_STEPS * DXDIM;
  float* Bf  = Af  + (size_t)T_STEPS * 256;
  float* AAb = Bf  + (size_t)T_STEPS * 256;
  float* BBb = AAb + (size_t)T_STEPS * 256;
  float* mub = BBb + (size_t)T_STEPS * 256;

  // h = relu(X @ W1 + b1): 8192x1024x1024, 128x256 block tiles, 64x64 wave tiles
  wmma_gemm<4, 4, 2, 4, true, false, false><<<dim3(64, 4), dim3(256), 0, stream>>>(
      X, X, W1, b1, h, T_STEPS, DXDIM, DXDIM);

  // mu = h @ Wmu + bmu: N=16, 512x16 block tiles
  wmma_gemm<4, 1, 8, 1, false, false, false><<<dim3(16, 1), dim3(256), 0, stream>>>(
      h, h, Wmu, bmu, mub, T_STEPS, DXDIM, DZDIM);

  // A = h @ WA + bA + I (identity folded into per-column bias): 256x128 block tiles
  wmma_gemm<4, 4, 4, 2, false, true, false><<<dim3(32, 2), dim3(256), 0, stream>>>(
      h, h, WA, bA, Af, T_STEPS, DXDIM, 256);

  // B = h[1:] @ WB[:1024] + h[:-1] @ WB[1024:] + bB   (8191 valid rows)
  wmma_gemm<4, 4, 4, 2, false, false, true><<<dim3(32, 2), dim3(256), 0, stream>>>(
      h + DXDIM, h, WB, bB, Bf, T_STEPS - 1, DXDIM, 256);

  // AA / BB block products via WMMA, one wave per t
  psd_kernel<<<dim3(T_STEPS / 8), dim3(256), 0, stream>>>(Af, Bf, AAb, BBb, T_STEPS);

  // Sequential Cholesky scan + entropy + sample_z
  scan_kernel<<<dim3(1), dim3(32), 0, stream>>>(AAb, BBb, mub, eps, out, T_STEPS);
}